// _QuantumQLSTMCell_65481071400690
// MI455X (gfx1250) — compile-verified
//
#include <hip/hip_runtime.h>
#include <math.h>

// D = A*B + C with one matrix per wave (wave32). fp32 WMMA 16x16x4.
typedef __attribute__((ext_vector_type(2))) float v2f;
typedef __attribute__((ext_vector_type(8))) float v8f;

#define NQ        8          // qubits == hidden dim
#define BATCH     4096
#define IN_DIM    128
#define FAN_IN    136        // 128 + 8
#define KSTEPS    34         // 136 / 4
#define ROWS_PB   64         // batch rows per block
#define CMB_ST    140        // LDS stride (dwords) for combined rows: mult of 4, 12*row%64 spreads banks
#define WT_ST     40         // LDS stride for W^T[136][32]: 2*40%64==16 -> halves hit disjoint banks
#define ANG_ST    33         // LDS stride for angle tile [64][32]

// Fast transcendentals: args are bounded here (|phi| ~< 6, |c| ~< 8), so
// v_exp_f32 / v_cos_f32 (~1e-6 rel err) are safe vs fp32 reference tolerance.
__device__ __forceinline__ float sig_(float v)   { return 1.0f / (1.0f + __expf(-v)); }
__device__ __forceinline__ float tanh_(float v)  { return 1.0f - 2.0f / (1.0f + __expf(2.0f * v)); }

__global__ __launch_bounds__(256)
void qlstm_fused_kernel(const float* __restrict__ x,  const float* __restrict__ h_prev,
                        const float* __restrict__ c_prev,
                        const float* __restrict__ Wf, const float* __restrict__ bf,
                        const float* __restrict__ Wi, const float* __restrict__ bi,
                        const float* __restrict__ Wu, const float* __restrict__ bu,
                        const float* __restrict__ Wo, const float* __restrict__ bo,
                        const float* __restrict__ tf, const float* __restrict__ ti,
                        const float* __restrict__ tu, const float* __restrict__ to_,
                        float* __restrict__ out)
{
    __shared__ __align__(16) float cmb[ROWS_PB * CMB_ST];  // 35840 B  combined rows
    __shared__ float             wt [FAN_IN  * WT_ST ];    // 21760 B  W^T [k][gate*8+n]
    __shared__ float             ang[ROWS_PB * ANG_ST];    //  8448 B  phi = W.c + b + theta

    const int tid  = threadIdx.x;
    const int row0 = blockIdx.x * ROWS_PB;

    // ---- stage combined = [x | h_prev] into LDS (coalesced) ----
    {
        const float4* x4 = (const float4*)x;               // 32 float4 per row
        for (int idx = tid; idx < ROWS_PB * (IN_DIM / 4); idx += 256) {
            int r = idx >> 5, c4 = idx & 31;
            float4 v = x4[(size_t)(row0 + r) * (IN_DIM / 4) + c4];
            *(float4*)&cmb[r * CMB_ST + c4 * 4] = v;
        }
        for (int idx = tid; idx < ROWS_PB * NQ; idx += 256) {
            int r = idx >> 3, c = idx & 7;
            cmb[r * CMB_ST + IN_DIM + c] = h_prev[(size_t)(row0 + r) * NQ + c];
        }
        // ---- stage weights transposed: wt[k][g*8+n] = W_g[n][k] ----
        for (int g = 0; g < 4; ++g) {
            const float* W = (g == 0) ? Wf : (g == 1) ? Wi : (g == 2) ? Wu : Wo;
            for (int idx = tid; idx < NQ * FAN_IN; idx += 256) {
                int n = idx / FAN_IN, k = idx - n * FAN_IN;
                wt[k * WT_ST + g * NQ + n] = W[idx];
            }
        }
    }
    __syncthreads();

    // ---- WMMA: each wave computes a 16(M) x 16(N) tile of phi = combined @ W^T + (b + theta) ----
    const int lane = tid & 31;
    const int wid  = tid >> 5;
    const int mt   = wid & 3;          // 4 M-tiles of 16 rows
    const int nt   = wid >> 2;         // 2 N-tiles of 16 cols
    const int m    = lane & 15;
    const int half = lane >> 4;
    const int ncol = nt * 16 + m;      // global column 0..31 (gate*8 + wire)

    // fold bias + theta into C init (column-constant)
    const int   gg   = ncol >> 3, wq = ncol & 7;
    const float* bp  = (gg == 0) ? bf : (gg == 1) ? bi : (gg == 2) ? bu : bo;
    const float* tp  = (gg == 0) ? tf : (gg == 1) ? ti : (gg == 2) ? tu : to_;
    const float binit = bp[wq] + tp[wq];

    v8f acc;
#pragma unroll
    for (int r = 0; r < 8; ++r) acc[r] = binit;

    // A frag: lane -> A[m][k0 + 2*half + r]  (ISA 32-bit 16x4 layout); consecutive -> b64 LDS load
    const float* arow = &cmb[(mt * 16 + m) * CMB_ST + 2 * half];
    // B frag: lane -> B[k0 + 2*half + r][ncol] (K split mirrors A/C layouts)
    const float* bcol = &wt[(2 * half) * WT_ST + ncol];

#pragma unroll
    for (int k0 = 0; k0 < KSTEPS; ++k0) {
        v2f a = *(const v2f*)(arow + 4 * k0);
        v2f b;
        b.x = bcol[(4 * k0 + 0) * WT_ST];
        b.y = bcol[(4 * k0 + 1) * WT_ST];
        acc = __builtin_amdgcn_wmma_f32_16x16x4_f32(false, a, false, b,
                                                    (short)0, acc, false, false);
    }

    // D layout: vgpr r, lane -> row = r + 8*half, col = m
#pragma unroll
    for (int r = 0; r < 8; ++r)
        ang[(mt * 16 + r + 8 * half) * ANG_ST + ncol] = acc[r];
    __syncthreads();

    // ---- epilogue: <Z_w> = prod_{v<=w} cos(phi_v); LSTM combine; one thread per row ----
    if (tid < ROWS_PB) {
        const int grow = row0 + tid;
        float gv[4][NQ];
#pragma unroll
        for (int g = 0; g < 4; ++g) {
            float p = 1.0f;
#pragma unroll
            for (int w = 0; w < NQ; ++w) {
                p *= __cosf(ang[tid * ANG_ST + g * NQ + w]);
                gv[g][w] = p;
            }
        }
#pragma unroll
        for (int w = 0; w < NQ; ++w) {
            float f = sig_(gv[0][w]);
            float i = sig_(gv[1][w]);
            float u = tanh_(sig_(gv[2][w]));
            float o = sig_(gv[3][w]);
            float cv = f * c_prev[(size_t)grow * NQ + w] + i * u;
            out[(size_t)grow * NQ + w]                      = o * tanh_(cv); // h
            out[(size_t)BATCH * NQ + (size_t)grow * NQ + w] = cv;            // c
        }
    }
}

extern "C" void kernel_launch(void* const* d_in, const int* in_sizes, int n_in,
                              void* d_out, int out_size, void* d_ws, size_t ws_size,
                              hipStream_t stream) {
    (void)in_sizes; (void)n_in; (void)out_size; (void)d_ws; (void)ws_size;
    const float* x      = (const float*)d_in[0];
    const float* h_prev = (const float*)d_in[1];
    const float* c_prev = (const float*)d_in[2];
    const float* Wf = (const float*)d_in[3];  const float* bf = (const float*)d_in[4];
    const float* Wi = (const float*)d_in[5];  const float* bi = (const float*)d_in[6];
    const float* Wu = (const float*)d_in[7];  const float* bu = (const float*)d_in[8];
    const float* Wo = (const float*)d_in[9];  const float* bo = (const float*)d_in[10];
    const float* tf = (const float*)d_in[11]; const float* ti = (const float*)d_in[12];
    const float* tu = (const float*)d_in[13]; const float* to_ = (const float*)d_in[14];
    float* out = (float*)d_out;

    dim3 grid(BATCH / ROWS_PB);   // 64 blocks
    dim3 block(256);              // 8 wave32s -> 4 M-tiles x 2 N-tiles
    qlstm_fused_kernel<<<grid, block, 0, stream>>>(x, h_prev, c_prev,
                                                   Wf, bf, Wi, bi, Wu, bu, Wo, bo,
                                                   tf, ti, tu, to_, out);
}